// YOlOLoss_7748121002658
// MI455X (gfx1250) — compile-verified
//
#include <hip/hip_runtime.h>
#include <math.h>

// ---- problem constants (from reference) ----
#define BSZ   16
#define TN    50
#define AA    3
#define CC    80
#define HH    76
#define WW    76
#define BAC   85                 // 5 + C channels per anchor
#define HW    (HH * WW)          // 5776
#define NCELL (BSZ * AA * HH * WW) // 277248

typedef float v2f __attribute__((ext_vector_type(2)));
typedef float v8f __attribute__((ext_vector_type(8)));

// ---------------------------------------------------------------------------
// Wave32 horizontal sum via V_WMMA_F32_16X16X4_F32 against an all-ones B.
// A-layout (16x4 f32): lane L VGPR0 = A[L%16, L<16 ? 0 : 2]; VGPR1 = K=1/3.
// With B==ones: D[m,n] = v[m] + v[m+16] for all n.
// D-layout: lane j<16 VGPR r = D[r, j]; lane j>=16 VGPR r = D[r+8, j-16].
// sum of 8 D VGPRs per lane + xor-16 shuffle => total over all 32 lanes.
// EXEC must be all-ones here (callers invoke it in convergent code).
// ---------------------------------------------------------------------------
__device__ __forceinline__ float wave_reduce_sum_wmma(float v) {
  v2f a; a[0] = v;    a[1] = 0.0f;
  v2f b; b[0] = 1.0f; b[1] = 1.0f;
  v8f c = {0.f, 0.f, 0.f, 0.f, 0.f, 0.f, 0.f, 0.f};
  v8f d = __builtin_amdgcn_wmma_f32_16x16x4_f32(
      /*neg_a=*/false, a, /*neg_b=*/false, b,
      /*c_mod=*/(short)0, c, /*reuse_a=*/false, /*reuse_b=*/false);
  float s = d[0] + d[1] + d[2] + d[3] + d[4] + d[5] + d[6] + d[7];
  s += __shfl_xor(s, 16, 32);
  return s;
}

__device__ __forceinline__ float sigmoidf_(float z) {
  return 1.0f / (1.0f + expf(-z));
}

// ---------------------------------------------------------------------------
// Kernel 1: init workspace grids (d_ws is poisoned 0xAA by the harness).
// obj=0, noobj=1, class-bitmask=0, accumulators=0.
// ---------------------------------------------------------------------------
__global__ void __launch_bounds__(256)
k_yolo_init(unsigned char* __restrict__ obj, unsigned char* __restrict__ noobj,
            unsigned int* __restrict__ clsbits, float* __restrict__ acc) {
  const int tid = blockIdx.x * blockDim.x + threadIdx.x;
  const int stride = gridDim.x * blockDim.x;
  if (tid < 8) acc[tid] = 0.0f;
  for (int i = tid; i < NCELL; i += stride) { obj[i] = 0; noobj[i] = 1; }
  for (int i = tid; i < 3 * NCELL; i += stride) clsbits[i] = 0u;
}

// ---------------------------------------------------------------------------
// Kernel 2: build_targets. One thread per batch (batches never collide);
// sequential over t preserves the reference scan's last-write-wins order.
// ---------------------------------------------------------------------------
__global__ void k_yolo_build(const float* __restrict__ tgt,
                             unsigned char* __restrict__ obj,
                             unsigned char* __restrict__ noobj,
                             float* __restrict__ tx, float* __restrict__ ty,
                             float* __restrict__ tw, float* __restrict__ th,
                             unsigned int* __restrict__ clsbits) {
  const int b = threadIdx.x;
  if (b >= BSZ) return;
  const float aw[3] = {1.25f, 2.0f, 4.125f};   // ANCHORS / (608/76)
  const float ah[3] = {1.625f, 3.75f, 2.875f};

  for (int t = 0; t < TN; ++t) {
    const float* s = tgt + (size_t)(b * TN + t) * 5;
    const float gx = s[0] * (float)WW;
    const float gy = s[1] * (float)HH;
    const float gw = s[2] * (float)WW;
    const float gh = s[3] * (float)HH;
    const int   gc = (int)s[4];
    const bool valid = (gx + gy + gw + gh) != 0.0f;
    if (!valid) continue;                      // all updates are v-gated

    int gi = (int)floorf(gx); if (gi == 0) gi = 1;
    int gj = (int)floorf(gy); if (gj == 0) gj = 1;

    float iou[3]; int best = 0; float best_iou = -1.0f;
    for (int a = 0; a < 3; ++a) {
      const float inter = fminf(gw, aw[a]) * fminf(gh, ah[a]);
      const float uni   = gw * gh + 1e-16f + aw[a] * ah[a] - inter;
      iou[a] = inter / uni;
      if (iou[a] > best_iou) { best_iou = iou[a]; best = a; }  // first-max, like argmax
    }

    // faithful to reference quirk: ignore-clear indexes noobj[b, :, gi, gj]
    // (gi on the H axis, gj on the W axis — transposed vs the other updates)
    for (int a = 0; a < 3; ++a)
      if (iou[a] > 0.5f)
        noobj[((b * AA + a) * HH + gi) * WW + gj] = 0;

    const int idx = ((b * AA + best) * HH + gj) * WW + gi;
    obj[idx] = 1;
    noobj[idx] = 0;
    tx[idx] = gx - (float)gi;
    ty[idx] = gy - (float)gj;
    tw[idx] = logf(gw / aw[best] + 1e-16f);
    th[idx] = logf(gh / ah[best] + 1e-16f);
    clsbits[idx * 3 + (gc >> 5)] |= (1u << (gc & 31));  // one-hot bits accumulate
  }
}

// ---------------------------------------------------------------------------
// Kernel 3: masked reductions. Only the conf channel is read for noobj cells;
// the other 84 channels are touched only at the ~800 obj cells.
// Wave-level reduction uses V_WMMA_F32_16X16X4_F32; one f32 atomic per
// accumulator per wave.
// acc: [0]=n_obj [1]=n_noobj [2]=obj terms [3]=noobj conf BCE [4]=cls BCE
// ---------------------------------------------------------------------------
__global__ void __launch_bounds__(256)
k_yolo_loss(const float* __restrict__ pred,
            const unsigned char* __restrict__ obj,
            const unsigned char* __restrict__ noobj,
            const float* __restrict__ tx, const float* __restrict__ ty,
            const float* __restrict__ tw, const float* __restrict__ th,
            const unsigned int* __restrict__ clsbits,
            float* __restrict__ acc) {
  float cObj = 0.f, cNoobj = 0.f, sObj = 0.f, sNoobj = 0.f, sCls = 0.f;
  const int stride = gridDim.x * blockDim.x;

  for (int idx = blockIdx.x * blockDim.x + threadIdx.x; idx < NCELL; idx += stride) {
    const int j = idx % WW;
    int r = idx / WW;
    const int i = r % HH; r /= HH;
    const int a = r % AA;
    const int b = r / AA;
    // pred is (B, A*85, H, W); channel ch of anchor a => base + ch*HW
    const float* pb = pred + ((size_t)(b * AA + a) * BAC) * HW + (size_t)i * WW + j;

    const unsigned char o  = obj[idx];
    const unsigned char nb = noobj[idx];

    if (nb) {  // loss_conf_noobj: target tconf==0 on noobj cells
      cNoobj += 1.0f;
      const float p = sigmoidf_(pb[4 * HW]);
      sNoobj += -fmaxf(log1pf(-p), -100.0f);
    }
    if (o) {
      cObj += 1.0f;
      const float px = sigmoidf_(pb[0]);
      const float py = sigmoidf_(pb[HW]);
      const float w_ = pb[2 * HW];
      const float h_ = pb[3 * HW];
      const float pc = sigmoidf_(pb[4 * HW]);
      const float dx = px - tx[idx], dy = py - ty[idx];
      const float dw = w_ - tw[idx], dh = h_ - th[idx];
      sObj += dx * dx + dy * dy + dw * dw + dh * dh;       // x,y,w,h MSE
      sObj += -fmaxf(logf(pc), -100.0f);                    // conf BCE, t=1
      const unsigned int m0 = clsbits[idx * 3 + 0];
      const unsigned int m1 = clsbits[idx * 3 + 1];
      const unsigned int m2 = clsbits[idx * 3 + 2];
      for (int c = 0; c < CC; ++c) {
        const float q = sigmoidf_(pb[(5 + c) * HW]);
        const unsigned int m = (c < 32) ? m0 : (c < 64) ? m1 : m2;
        const bool tset = (m >> (c & 31)) & 1u;
        sCls += tset ? -fmaxf(logf(q), -100.0f)
                     : -fmaxf(log1pf(-q), -100.0f);
      }
    }
  }

  // convergent: every lane reaches here with full EXEC (WMMA requirement)
  cObj   = wave_reduce_sum_wmma(cObj);
  cNoobj = wave_reduce_sum_wmma(cNoobj);
  sObj   = wave_reduce_sum_wmma(sObj);
  sNoobj = wave_reduce_sum_wmma(sNoobj);
  sCls   = wave_reduce_sum_wmma(sCls);

  if ((threadIdx.x & 31) == 0) {
    atomicAdd(&acc[0], cObj);
    atomicAdd(&acc[1], cNoobj);
    atomicAdd(&acc[2], sObj);
    atomicAdd(&acc[3], sNoobj);
    atomicAdd(&acc[4], sCls);
  }
}

// ---------------------------------------------------------------------------
// Kernel 4: finalize scalar loss.
// ---------------------------------------------------------------------------
__global__ void k_yolo_final(const float* __restrict__ acc, float* __restrict__ out) {
  if (blockIdx.x == 0 && threadIdx.x == 0) {
    const float nObj   = fmaxf(acc[0], 1.0f);
    const float nNoobj = fmaxf(acc[1], 1.0f);
    out[0] = acc[2] / nObj                       // x+y+w+h MSE + conf_obj BCE (OBJ_SCALE=1)
           + 100.0f * acc[3] / nNoobj            // NOOBJ_SCALE * conf_noobj BCE
           + acc[4] / (nObj * (float)CC);        // class BCE
  }
}

// ---------------------------------------------------------------------------
extern "C" void kernel_launch(void* const* d_in, const int* in_sizes, int n_in,
                              void* d_out, int out_size, void* d_ws, size_t ws_size,
                              hipStream_t stream) {
  (void)in_sizes; (void)n_in; (void)out_size; (void)ws_size;
  const float* pred = (const float*)d_in[0];   // (16, 255, 76, 76) f32
  const float* tgts = (const float*)d_in[1];   // (16, 50, 5) f32
  float* out = (float*)d_out;                  // scalar loss

  // workspace layout (~8.4 MB):
  //   [0,32)                      : 8 f32 accumulators
  //   [32, 32+N)                  : obj   (u8)
  //   [.., +N)                    : noobj (u8)
  //   then tx, ty, tw, th (f32 each), clsbits (3*u32 per cell)
  char* ws = (char*)d_ws;
  float* acc = (float*)ws;
  unsigned char* obj   = (unsigned char*)(ws + 32);
  unsigned char* noobj = obj + NCELL;
  float* tx = (float*)(ws + 32 + 2 * (size_t)NCELL);
  float* ty = tx + NCELL;
  float* tw = ty + NCELL;
  float* th = tw + NCELL;
  unsigned int* clsbits = (unsigned int*)(th + NCELL);

  k_yolo_init <<<512, 256, 0, stream>>>(obj, noobj, clsbits, acc);
  k_yolo_build<<<1, 32, 0, stream>>>(tgts, obj, noobj, tx, ty, tw, th, clsbits);
  k_yolo_loss <<<512, 256, 0, stream>>>(pred, obj, noobj, tx, ty, tw, th, clsbits, acc);
  k_yolo_final<<<1, 1, 0, stream>>>(acc, out);
}